// DiagonalSSMKernel_56118042689803
// MI455X (gfx1250) — compile-verified
//
#include <hip/hip_runtime.h>
#include <hip/hip_bf16.h>
#include <math.h>

typedef __attribute__((ext_vector_type(2))) float v2f;
typedef __attribute__((ext_vector_type(8))) float v8f;

#define HD     1024   // d_model
#define NP     32     // poles
#define LTOT   4096   // sequence length
#define LO     64     // outer time factor
#define LI     64     // inner time factor
#define AS     68     // A LDS row stride (pad: banks (4m+k)%64 conflict-free)
#define BS     80     // B LDS row stride (pad: halves in disjoint bank sets)

__launch_bounds__(256)
__global__ void s4d_vandermonde_wmma(const float* __restrict__ log_dt,
                                     const float* __restrict__ log_a_real,
                                     const float* __restrict__ a_imag,
                                     const float* __restrict__ coeffs,
                                     float* __restrict__ out)
{
    __shared__ float Bs[64 * BS];    // [k=0..63][j=0..63]: rows 0..31 Re(w^j), 32..63 Im(w^j)
    __shared__ float As[128 * AS];   // [m=d*64+l0][k]: cols 0..31 Re(sc*w64^l0), 32..63 -Im(...)
    __shared__ float wdr[NP], wdi[NP];
    __shared__ float scr[2][NP], sci[2][NP];

    const int h   = blockIdx.x;
    const int tid = threadIdx.x;

    // ---- Phase 1: per-(h,n) scalars: delta_a and scaled_c (x2 folded in) ----
    if (tid < NP) {
        const int n = tid;
        float dt = expf(log_dt[h]);
        float ar = -expf(log_a_real[h * NP + n]);
        float ai = a_imag[h * NP + n];
        float dr = ar * dt, di = ai * dt;          // delta_a
        wdr[n] = dr; wdi[n] = di;
        // (exp(delta_a) - 1) / a
        float e = expf(dr);
        float s, c;  sincosf(di, &s, &c);
        float exm1r = e * c - 1.0f;
        float exm1i = e * s;
        float inv   = 1.0f / (ar * ar + ai * ai);
        float fr = (exm1r * ar + exm1i * ai) * inv;
        float fi = (exm1i * ar - exm1r * ai) * inv;
        #pragma unroll
        for (int d = 0; d < 2; ++d) {
            float cr = coeffs[(((size_t)d * HD + h) * NP + n) * 2 + 0];
            float ci = coeffs[(((size_t)d * HD + h) * NP + n) * 2 + 1];
            scr[d][n] = 2.0f * (cr * fr - ci * fi);
            sci[d][n] = 2.0f * (cr * fi + ci * fr);
        }
    }
    __syncthreads();

    // ---- Phase 2: B = inner twiddles w^j, j = 0..63 ----
    for (int idx = tid; idx < NP * LI; idx += 256) {
        int n = idx >> 6, j = idx & 63;
        float t = (float)j;
        float mag = expf(wdr[n] * t);
        float s, c;  sincosf(wdi[n] * t, &s, &c);
        Bs[n * BS + j]        = mag * c;    // Re
        Bs[(NP + n) * BS + j] = mag * s;    // Im
    }

    // ---- Phase 3: A = scaled_c * (w^64)^l0, split Re / -Im ----
    for (int idx = tid; idx < 2 * LO * NP; idx += 256) {
        int n = idx & 31, l0 = (idx >> 5) & 63, d = idx >> 11;
        float t = 64.0f * (float)l0;
        float mag = expf(wdr[n] * t);
        float s, c;  sincosf(wdi[n] * t, &s, &c);
        float vr = mag * c, vi = mag * s;
        float Ar = scr[d][n] * vr - sci[d][n] * vi;
        float Ai = scr[d][n] * vi + sci[d][n] * vr;
        int m = d * LO + l0;
        As[m * AS + n]      = Ar;
        As[m * AS + NP + n] = -Ai;
    }
    __syncthreads();

    // ---- Phase 4: C[128,64] = A[128,64] x B[64,64] via f32 WMMA 16x16x4 ----
    const int wave = tid >> 5;          // 8 waves, one 16-row m-tile each
    const int lane = tid & 31;
    const int col  = lane & 15;
    const int half = lane >> 4;         // 0: K=0/1, 1: K=2/3 within each k-step
    const int mRow = wave * 16 + col;   // A-fragment row for this lane

    v8f acc[4];
    #pragma unroll
    for (int jt = 0; jt < 4; ++jt) {
        v8f c = {};
        #pragma unroll
        for (int ks = 0; ks < 16; ++ks) {
            const int kb = ks * 4 + half * 2;
            v2f a, b;
            a.x = As[mRow * AS + kb];
            a.y = As[mRow * AS + kb + 1];
            b.x = Bs[kb * BS + jt * 16 + col];
            b.y = Bs[(kb + 1) * BS + jt * 16 + col];
            c = __builtin_amdgcn_wmma_f32_16x16x4_f32(
                    false, a, false, b, (short)0, c, false, false);
        }
        acc[jt] = c;
    }

    // Write C tiles over this wave's own A rows (no cross-wave hazard).
    #pragma unroll
    for (int jt = 0; jt < 4; ++jt) {
        #pragma unroll
        for (int r = 0; r < 8; ++r) {
            int m = wave * 16 + half * 8 + r;   // C layout: VGPR r -> M=r / M=8+r
            int j = jt * 16 + col;
            As[m * AS + j] = acc[jt][r];
        }
    }
    __syncthreads();

    // ---- Phase 5: coalesced streaming store of kernels[d, h, l0*64 + j] ----
    for (int idx = tid; idx < 128 * 64; idx += 256) {
        int m = idx >> 6, j = idx & 63;
        int d = m >> 6, l0 = m & 63;
        out[((size_t)d * HD + h) * LTOT + (size_t)l0 * 64 + j] = As[m * AS + j];
    }
}

extern "C" void kernel_launch(void* const* d_in, const int* in_sizes, int n_in,
                              void* d_out, int out_size, void* d_ws, size_t ws_size,
                              hipStream_t stream) {
    (void)in_sizes; (void)n_in; (void)d_ws; (void)ws_size; (void)out_size;
    const float* log_dt     = (const float*)d_in[0];
    const float* log_a_real = (const float*)d_in[1];
    const float* a_imag     = (const float*)d_in[2];
    const float* coeffs     = (const float*)d_in[3];
    // d_in[4] = sequence_length (fixed 4096, baked into LTOT)
    float* out = (float*)d_out;

    s4d_vandermonde_wmma<<<HD, 256, 0, stream>>>(log_dt, log_a_real, a_imag, coeffs, out);
}